// MI_21998822490468
// MI455X (gfx1250) — compile-verified
//
#include <hip/hip_runtime.h>
#include <hip/hip_bf16.h>

// Mutual information of two int32 label images, labels in [0, 256).
// Pipeline:
//   1) per-WGP joint histogram in 256KB LDS (CDNA5: 320KB/WGP), ds_add_u32
//   2) merge partials (streaming stores + reduce kernel, or atomic fallback)
//   3) marginals via v_wmma_f32_16x16x4_f32 (J*ones, J^T*ones), then MI sum

#define NB        256
#define NB2       65536
#define HBLOCK    1024
#define NPART     128

typedef float v2f __attribute__((ext_vector_type(2)));
typedef float v8f __attribute__((ext_vector_type(8)));

// ---------------------------------------------------------------- zero kernel
__global__ void mi_zero_kernel(unsigned* __restrict__ hist) {
    int i = blockIdx.x * blockDim.x + threadIdx.x;
    if (i < NB2) hist[i] = 0u;
}

// ------------------------------------------------- LDS-private joint histogram
// Each 1024-thread block owns a full 256KB histogram in LDS (one WGP each).
__device__ __forceinline__ void mi_build_lds_hist(const int* __restrict__ a,
                                                  const int* __restrict__ b,
                                                  long long n,
                                                  unsigned* lhist) {
    for (int i = threadIdx.x; i < NB2; i += HBLOCK) lhist[i] = 0u;
    __syncthreads();

    const long long n4 = n >> 2;  // int4 vectorized body
    const int4* __restrict__ a4 = (const int4*)a;
    const int4* __restrict__ b4 = (const int4*)b;
    const long long stride = (long long)gridDim.x * HBLOCK;

    for (long long i = (long long)blockIdx.x * HBLOCK + threadIdx.x; i < n4; i += stride) {
        int4 av = a4[i];
        int4 bv = b4[i];
        atomicAdd(&lhist[(av.x << 8) | bv.x], 1u);
        atomicAdd(&lhist[(av.y << 8) | bv.y], 1u);
        atomicAdd(&lhist[(av.z << 8) | bv.z], 1u);
        atomicAdd(&lhist[(av.w << 8) | bv.w], 1u);
    }
    // scalar tail (n may not be a multiple of 4)
    for (long long i = (n4 << 2) + (long long)blockIdx.x * HBLOCK + threadIdx.x;
         i < n; i += stride) {
        atomicAdd(&lhist[(a[i] << 8) | b[i]], 1u);
    }
    __syncthreads();
}

// Variant A: stream private histogram to a per-block partial (plain b128 stores)
__global__ __launch_bounds__(HBLOCK)
void mi_hist_partial_kernel(const int* __restrict__ a, const int* __restrict__ b,
                            unsigned* __restrict__ partials, long long n) {
    __shared__ unsigned lhist[NB2];   // 256 KB of the WGP's 320 KB LDS
    mi_build_lds_hist(a, b, n, lhist);

    uint4* __restrict__ dst = (uint4*)(partials + (size_t)blockIdx.x * NB2);
    const uint4* __restrict__ src = (const uint4*)lhist;
    for (int i = threadIdx.x; i < NB2 / 4; i += HBLOCK) dst[i] = src[i];
}

// Variant B: atomic flush (used only if scratch is too small for partials)
__global__ __launch_bounds__(HBLOCK)
void mi_hist_atomic_kernel(const int* __restrict__ a, const int* __restrict__ b,
                           unsigned* __restrict__ ghist, long long n) {
    __shared__ unsigned lhist[NB2];
    mi_build_lds_hist(a, b, n, lhist);

    for (int i = threadIdx.x; i < NB2; i += HBLOCK) {
        unsigned v = lhist[i];
        if (v) atomicAdd(&ghist[i], v);
    }
}

// ------------------------------------------------------------- partial reduce
__global__ void mi_reduce_kernel(const unsigned* __restrict__ partials,
                                 unsigned* __restrict__ hist, int nparts) {
    int bin = blockIdx.x * blockDim.x + threadIdx.x;
    if (bin >= NB2) return;
    unsigned s = 0u;
    for (int p = 0; p < nparts; ++p) s += partials[(size_t)p * NB2 + bin];
    hist[bin] = s;
}

// --------------------------------------------------- marginals (WMMA) + MI sum
// 512 threads = 16 waves; wave w owns rows/cols [16w, 16w+16).
// pi = J * ones, pj = J^T * ones via v_wmma_f32_16x16x4_f32 (exact: counts < 2^24).
__global__ __launch_bounds__(512)
void mi_final_kernel(const unsigned* __restrict__ hist, float* __restrict__ out,
                     float fn) {
    __shared__ float slpi[NB];
    __shared__ float slpj[NB];
    __shared__ double sred[512];

    const int tid  = threadIdx.x;
    const int wave = tid >> 5;
    const int lane = tid & 31;
    const int m    = lane & 15;              // row (or col) within the 16-stripe
    const int kh   = (lane < 16) ? 0 : 2;    // f32 A-tile: lanes 0-15 K={0,1}, 16-31 K={2,3}
    const int base = wave * 16;

    v2f ones; ones.x = 1.0f; ones.y = 1.0f;
    v8f accR = {};   // row sums  (pi stripe)
    v8f accC = {};   // col sums  (pj stripe)

    for (int k = 0; k < NB; k += 4) {
        const int kb = k + kh;
        v2f ar, ac;
        // A[m][kb..kb+1] of J for row sums
        ar.x = (float)hist[(base + m) * NB + kb];
        ar.y = (float)hist[(base + m) * NB + kb + 1];
        // A[m][kb..kb+1] of J^T for col sums
        ac.x = (float)hist[kb * NB + base + m];
        ac.y = (float)hist[(kb + 1) * NB + base + m];
        accR = __builtin_amdgcn_wmma_f32_16x16x4_f32(
            false, ar, false, ones, (short)0, accR, false, false);
        accC = __builtin_amdgcn_wmma_f32_16x16x4_f32(
            false, ac, false, ones, (short)0, accC, false, false);
    }

    // D layout: lane L, vgpr v -> M = v + (L<16 ? 0 : 8), N = L&15.
    // Every column of D holds the same sums (B == ones), so N=0 lanes suffice.
    if (lane == 0 || lane == 16) {
        const int moff = (lane == 0) ? 0 : 8;
        #pragma unroll
        for (int v = 0; v < 8; ++v) {
            float pi = accR[v];
            float pj = accC[v];
            slpi[base + moff + v] = __builtin_logf(pi > 0.f ? pi : 1.f);
            slpj[base + moff + v] = __builtin_logf(pj > 0.f ? pj : 1.f);
        }
    }
    __syncthreads();

    const float logN = __builtin_logf(fn);
    double acc = 0.0;
    for (int idx = tid; idx < NB2; idx += 512) {
        unsigned v = hist[idx];
        if (v) {
            const int i = idx >> 8;
            const int j = idx & 255;
            const float lv = __builtin_logf((float)v);
            acc += (double)((float)v * (lv - slpi[i] - slpj[j] + logN));
        }
    }
    sred[tid] = acc;
    __syncthreads();
    for (int s = 256; s > 0; s >>= 1) {
        if (tid < s) sred[tid] += sred[tid + s];
        __syncthreads();
    }
    if (tid == 0) {
        float mi = (float)(sred[0] / (double)fn);
        out[0] = mi > 0.f ? mi : 0.f;
    }
}

// ------------------------------------------------------------------- launcher
extern "C" void kernel_launch(void* const* d_in, const int* in_sizes, int n_in,
                              void* d_out, int out_size, void* d_ws, size_t ws_size,
                              hipStream_t stream) {
    const int* a = (const int*)d_in[0];
    const int* b = (const int*)d_in[1];
    float* out   = (float*)d_out;
    const long long n = (long long)in_sizes[0];
    unsigned* ws = (unsigned*)d_ws;

    const size_t need_partial = ((size_t)NPART + 1) * NB2 * sizeof(unsigned);

    if (ws_size >= need_partial) {
        // Preferred path: per-block partial histograms, zero global atomics.
        unsigned* partials = ws;
        unsigned* hist     = ws + (size_t)NPART * NB2;
        mi_hist_partial_kernel<<<NPART, HBLOCK, 0, stream>>>(a, b, partials, n);
        mi_reduce_kernel<<<NB2 / 256, 256, 0, stream>>>(partials, hist, NPART);
        mi_final_kernel<<<1, 512, 0, stream>>>(hist, out, (float)n);
    } else {
        // Fallback: single global histogram with zero-skipping atomic flush.
        unsigned* hist = ws;
        mi_zero_kernel<<<(NB2 + HBLOCK - 1) / HBLOCK, HBLOCK, 0, stream>>>(hist);
        mi_hist_atomic_kernel<<<NPART, HBLOCK, 0, stream>>>(a, b, hist, n);
        mi_final_kernel<<<1, 512, 0, stream>>>(hist, out, (float)n);
    }
}